// ContactMapHead_9500467659177
// MI455X (gfx1250) — compile-verified
//
#include <hip/hip_runtime.h>
#include <math.h>
#include <stdint.h>

#define NB 2
#define NL 401
#define ND 128
#define LJ 32
#define BIROWS (NB * NL)            // 802
#define MTILES ((BIROWS + 15) / 16) // 51

typedef float v2f __attribute__((ext_vector_type(2)));
typedef float v8f __attribute__((ext_vector_type(8)));
#define Z8 {0.f, 0.f, 0.f, 0.f, 0.f, 0.f, 0.f, 0.f}

__device__ __forceinline__ v8f wmma4(v2f a, v2f b, v8f c) {
  // V_WMMA_F32_16X16X4_F32 : D = A(16x4) * B(4x16) + C(16x16)
  return __builtin_amdgcn_wmma_f32_16x16x4_f32(false, a, false, b, (short)0, c,
                                               false, false);
}

__device__ __forceinline__ float gelu_exact(float v) {
  return 0.5f * v * (1.0f + erff(v * 0.70710678118654752f));
}

// CDNA5 async global->LDS copy (16B per lane), tracked by ASYNCcnt.
__device__ __forceinline__ void async_b128(float* lds_ptr, const float* gptr) {
  uint32_t l = (uint32_t)(uintptr_t)lds_ptr;  // low 32 bits of generic = LDS offset
  asm volatile("global_load_async_to_lds_b128 %0, %1, off"
               :: "v"(l), "v"(gptr) : "memory");
}
__device__ __forceinline__ void async_wait0() {
  asm volatile("s_wait_asynccnt 0x0" ::: "memory");
}

// ---------------------------------------------------------------------------
// K1: T[bi, k*128+e] = sum_d x[bi, d] * W_bil[k, d, e]
// grid = (51, 128), block = 256 (8 waves). Each wave: one 16x16 C tile.
// ---------------------------------------------------------------------------
__global__ __launch_bounds__(256) void k_bilinear_T(const float* __restrict__ x,
                                                    const float* __restrict__ Wb,
                                                    float* __restrict__ T) {
  const int lane = threadIdx.x & 31;
  const int wave = threadIdx.x >> 5;
  const int half = lane >> 4;
  const int l16  = lane & 15;
  const int c2   = half << 1;

  const int m0 = blockIdx.x * 16;
  const int n  = blockIdx.y * 128 + wave * 16 + l16;  // output column (k,e)
  const int k  = n >> 7;
  const int e  = n & 127;

  int arow = m0 + l16;
  if (arow > BIROWS - 1) arow = BIROWS - 1;           // clamp (rows >=802 not stored)

  v8f c = Z8;
  for (int d0 = 0; d0 < ND; d0 += 4) {
    v2f a = *(const v2f*)(&x[(size_t)arow * ND + d0 + c2]);
    v2f bv;
    bv[0] = Wb[(size_t)k * (ND * ND) + (size_t)(d0 + c2) * ND + e];
    bv[1] = Wb[(size_t)k * (ND * ND) + (size_t)(d0 + c2 + 1) * ND + e];
    c = wmma4(a, bv, c);
  }
#pragma unroll
  for (int r = 0; r < 8; ++r) {
    int mr = m0 + r + (half << 3);
    if (mr < BIROWS) T[(size_t)mr * (ND * ND) + n] = c[r];
  }
}

// ---------------------------------------------------------------------------
// K2: fused  pair -> +b_bil -> LayerNorm -> @W1^T+b1 -> GELU -> dot w2 -> +b2
// One block per (b,i). 256 threads (8 waves).
// ---------------------------------------------------------------------------
__global__ __launch_bounds__(256) void k_fused(const float* __restrict__ x,
                                               const float* __restrict__ T,
                                               const float* __restrict__ b_bil,
                                               const float* __restrict__ ln_g,
                                               const float* __restrict__ ln_b,
                                               const float* __restrict__ W1,
                                               const float* __restrict__ b1,
                                               const float* __restrict__ w2,
                                               const float* __restrict__ b2,
                                               float* __restrict__ contact) {
  __shared__ float sTi[ND * 132];   // T[b,i,k,e], padded stride
  __shared__ float sX [LJ * 132];   // x[b, j0..j0+31, e]
  __shared__ float sP [LJ * 132];   // pair tile [j][k] -> LN'd in place
  __shared__ float sC [LJ];         // per-j contact accumulator
  __shared__ float sBb[ND], sG[ND], sBe[ND], sB1[ND], sW2[ND];

  const int tid  = threadIdx.x;
  const int lane = tid & 31;
  const int wave = tid >> 5;
  const int half = lane >> 4;
  const int l16  = lane & 15;
  const int c2   = half << 1;

  const int bi = blockIdx.x;
  const int b  = bi / NL;
  const int i  = bi % NL;

  // ---- async-stage T[b,i] into LDS: 4096 x b128 chunks, 16 per thread ----
  {
    const float* gT = T + (size_t)bi * (ND * ND);
#pragma unroll
    for (int it = 0; it < 16; ++it) {
      int idx = tid + it * 256;        // chunk id, 32 chunks per 128-float row
      int k  = idx >> 5;
      int cr = (idx & 31) << 2;        // float offset within row
      async_b128(&sTi[k * 132 + cr], gT + k * ND + cr);
    }
  }
  if (tid < ND) {
    sBb[tid] = b_bil[tid]; sG[tid] = ln_g[tid]; sBe[tid] = ln_b[tid];
    sB1[tid] = b1[tid];    sW2[tid] = w2[tid];
  }
  async_wait0();
  __syncthreads();

  const float bias2 = b2[0];

  for (int j0 = 0; j0 < NL; j0 += LJ) {
    // ---- async-stage X tile (row index clamped; tail cols never stored) ----
    {
      const float* gX = x + (size_t)b * NL * ND;
#pragma unroll
      for (int it = 0; it < 4; ++it) {
        int idx = tid + it * 256;      // 1024 chunks: 32 per row, 32 rows
        int jj = idx >> 5;
        int cr = (idx & 31) << 2;
        int j  = j0 + jj; if (j > NL - 1) j = NL - 1;
        async_b128(&sX[jj * 132 + cr], gX + (size_t)j * ND + cr);
      }
    }
    if (tid < LJ) sC[tid] = 0.f;
    async_wait0();
    __syncthreads();

    // ---- GEMM1: P[k, j] = sum_e T[k,e] * X[j,e]  (A reused for 2 N-tiles) ----
    {
      const int k0 = wave * 16;        // wave owns a 16-row k strip
      v8f c0 = Z8, c1 = Z8;
      for (int e0 = 0; e0 < ND; e0 += 4) {
        v2f a   = *(const v2f*)(&sTi[(k0 + l16) * 132 + e0 + c2]);
        v2f bv0 = *(const v2f*)(&sX [(l16)      * 132 + e0 + c2]);
        v2f bv1 = *(const v2f*)(&sX [(16 + l16) * 132 + e0 + c2]);
        c0 = wmma4(a, bv0, c0);
        c1 = wmma4(a, bv1, c1);
      }
#pragma unroll
      for (int r = 0; r < 8; ++r) {
        sP[(l16)      * 132 + k0 + r + (half << 3)] = c0[r];
        sP[(16 + l16) * 132 + k0 + r + (half << 3)] = c1[r];
      }
    }
    __syncthreads();

    // ---- +b_bil and LayerNorm over k, in place (8 threads per j row) ----
    {
      const int jj  = tid >> 3;
      const int sub = tid & 7;
      float s = 0.f, s2 = 0.f;
#pragma unroll
      for (int t = 0; t < 16; ++t) {
        int kk = sub * 16 + t;
        float v = sP[jj * 132 + kk] + sBb[kk];
        s += v; s2 += v * v;
      }
#pragma unroll
      for (int off = 1; off < 8; off <<= 1) {
        s  += __shfl_xor(s,  off, 32);
        s2 += __shfl_xor(s2, off, 32);
      }
      float mu   = s * (1.f / 128.f);
      float var  = s2 * (1.f / 128.f) - mu * mu;
      float rstd = rsqrtf(var + 1e-5f);
#pragma unroll
      for (int t = 0; t < 16; ++t) {
        int kk = sub * 16 + t;
        float v = sP[jj * 132 + kk] + sBb[kk];
        sP[jj * 132 + kk] = (v - mu) * rstd * sG[kk] + sBe[kk];
      }
    }
    __syncthreads();

    // ---- GEMM2: H2[j,e] = sum_k H[j,k]*W1[e,k]; B reused for 2 M-tiles ----
    {
      const int e0 = wave * 16;        // wave owns a 16-col e strip
      const int e  = e0 + l16;
      const float w2e = sW2[e];
      const float b1e = sB1[e];
      v8f c0 = Z8, c1 = Z8;
      for (int k0 = 0; k0 < ND; k0 += 4) {
        v2f bv = *(const v2f*)(&W1[(size_t)e * ND + k0 + c2]); // contiguous pair
        v2f a0 = *(const v2f*)(&sP[(l16)      * 132 + k0 + c2]);
        v2f a1 = *(const v2f*)(&sP[(16 + l16) * 132 + k0 + c2]);
        c0 = wmma4(a0, bv, c0);
        c1 = wmma4(a1, bv, c1);
      }
      // epilogue: GELU + *w2, cross-lane (16) reduce, LDS atomic accumulate
#pragma unroll
      for (int m = 0; m < 2; ++m) {
        v8f cc = (m == 0) ? c0 : c1;
        float pr[8];
#pragma unroll
        for (int r = 0; r < 8; ++r) pr[r] = gelu_exact(cc[r] + b1e) * w2e;
#pragma unroll
        for (int off = 1; off < 16; off <<= 1) {
#pragma unroll
          for (int r = 0; r < 8; ++r) pr[r] += __shfl_xor(pr[r], off, 32);
        }
        if (l16 == 0) {
          int base = m * 16 + (half << 3);
#pragma unroll
          for (int r = 0; r < 8; ++r) atomicAdd(&sC[base + r], pr[r]);
        }
      }
    }
    __syncthreads();

    if (tid < LJ) {
      int j = j0 + tid;
      if (j < NL)
        contact[((size_t)b * NL + i) * NL + j] = sC[tid] + bias2;
    }
    __syncthreads();
  }
}

// ---------------------------------------------------------------------------
// K3: in-place symmetrization. Pair {i,j} (i<=j) owned by exactly one thread.
// ---------------------------------------------------------------------------
__global__ __launch_bounds__(512) void k_sym(float* __restrict__ c) {
  const int i = blockIdx.x;
  const int b = blockIdx.y;
  const int j = threadIdx.x;
  if (j < NL && j >= i) {
    size_t ij = ((size_t)b * NL + i) * NL + j;
    size_t ji = ((size_t)b * NL + j) * NL + i;
    float m = 0.5f * (c[ij] + c[ji]);
    c[ij] = m;
    c[ji] = m;
  }
}

extern "C" void kernel_launch(void* const* d_in, const int* in_sizes, int n_in,
                              void* d_out, int out_size, void* d_ws, size_t ws_size,
                              hipStream_t stream) {
  const float* x    = (const float*)d_in[0];
  const float* Wbil = (const float*)d_in[1];
  const float* bbil = (const float*)d_in[2];
  const float* lng  = (const float*)d_in[3];
  const float* lnb  = (const float*)d_in[4];
  const float* W1   = (const float*)d_in[5];
  const float* b1   = (const float*)d_in[6];
  const float* w2   = (const float*)d_in[7];
  const float* b2   = (const float*)d_in[8];
  float* out = (float*)d_out;
  float* T   = (float*)d_ws;   // 802*128*128 f32 = 52.6 MB

  k_bilinear_T<<<dim3(MTILES, 128), 256, 0, stream>>>(x, Wbil, T);
  k_fused<<<BIROWS, 256, 0, stream>>>(x, T, bbil, lng, lnb, W1, b1, w2, b2, out);
  k_sym<<<dim3(NL, NB), 512, 0, stream>>>(out);
}